// SelfEntropyLayerV2_9371618640651
// MI455X (gfx1250) — compile-verified
//
#include <hip/hip_runtime.h>

typedef __attribute__((ext_vector_type(2))) float v2f;
typedef __attribute__((ext_vector_type(8))) float v8f;

#define HH   512
#define WW   512
#define CC   3
#define TILE 16
#define PATCH 24      // TILE + 2*4 halo
#define EPITCH 25     // padded LDS pitch for e-patch (odd -> conflict-free column reads)
#define HPITCH 17     // padded LDS pitch for horizontal sums

// One wave (32 lanes) computes one 16x16 output tile of one (batch, channel) plane.
// Pass 1: Hs(24x16) = E(24x24) x Wh(24x16)   (A = data, B = banded ones)
// Pass 2: O(16x16)  = Wv(16x24) x Hs(24x16)  (A = banded ones, B = data)
// Both via chained V_WMMA_F32_16X16X4_F32 (K = 24 -> 6 steps).
__global__ __launch_bounds__(32)
void entropy9x9_wmma_kernel(const float* __restrict__ x, float* __restrict__ out) {
    const int lane = threadIdx.x;              // 0..31
    const int tilesX = WW / TILE;              // 32
    const int tx = blockIdx.x % tilesX;
    const int ty = blockIdx.x / tilesX;
    const int c  = blockIdx.y;
    const int b  = blockIdx.z;
    const int x0 = tx * TILE, y0 = ty * TILE;

    __shared__ float eT[PATCH * EPITCH];       // e = p*log2(p+eps), 24x24 halo patch
    __shared__ float hs[PATCH * HPITCH];       // horizontal 9-tap sums, 24x16

    const float* plane = x + (size_t)b * HH * WW * CC + c;

    // ---- Stage 1: compute e for the 24x24 patch (zero outside the image) ----
    #pragma unroll
    for (int t = 0; t < (PATCH * PATCH) / 32; ++t) {   // 576 / 32 = 18, no divergence
        int id = t * 32 + lane;
        int i = id / PATCH;
        int j = id - i * PATCH;
        int gy = y0 + i - 4;
        int gx = x0 + j - 4;
        float e = 0.0f;
        if ((unsigned)gy < (unsigned)HH && (unsigned)gx < (unsigned)WW) {
            float p = plane[((size_t)gy * WW + gx) * CC];
            e = p * __log2f(p + 1e-10f);
        }
        eT[i * EPITCH + j] = e;
    }
    __syncthreads();   // 1-wave WG: S_NOP; same-wave LDS ops are in-order anyway

    const int m  = lane & 15;     // row index (A/D) / column index (B) within 16
    const int hi = lane >> 4;     // lane half select
    const int ka = hi * 2;        // f32 K=4 layout: v0 holds K0 (lo lanes) / K2 (hi lanes)

    // ---- Pass 1 (horizontal), two row blocks covering rows 0..23 ----
    #pragma unroll
    for (int rb = 0; rb < 2; ++rb) {
        const int R = rb * 8;                      // rows R..R+15 of the patch
        v8f acc = {};
        #pragma unroll
        for (int q = 0; q < 6; ++q) {
            const int k0 = q * 4;
            v2f a, bm;
            // A chunk (16x4 of E): lane -> row R+m, K = k0+ka, k0+ka+1
            a.x = eT[(R + m) * EPITCH + k0 + ka];
            a.y = eT[(R + m) * EPITCH + k0 + ka + 1];
            // B chunk (4x16 band): Wh[k][n] = (n <= k && k <= n+8), n = m
            const int kb = k0 + ka;
            bm.x = (m <= kb     && kb     <= m + 8) ? 1.0f : 0.0f;
            bm.y = (m <= kb + 1 && kb + 1 <= m + 8) ? 1.0f : 0.0f;
            acc = __builtin_amdgcn_wmma_f32_16x16x4_f32(false, a, false, bm,
                                                        (short)0, acc, false, false);
        }
        // D layout -> LDS: VGPR g holds rows R+g (lo lanes) / R+g+8 (hi lanes), col m
        #pragma unroll
        for (int g = 0; g < 8; ++g)
            hs[(R + g + hi * 8) * HPITCH + m] = acc[g];
    }
    __syncthreads();

    // ---- Pass 2 (vertical): O = Wv x Hs ----
    v8f acc = {};
    #pragma unroll
    for (int q = 0; q < 6; ++q) {
        const int k0 = q * 4;
        const int kk = k0 + ka;
        v2f a, bm;
        // A chunk (16x4 band): Wv[m][k] = (m <= k && k <= m+8)
        a.x = (m <= kk     && kk     <= m + 8) ? 1.0f : 0.0f;
        a.y = (m <= kk + 1 && kk + 1 <= m + 8) ? 1.0f : 0.0f;
        // B chunk (4x16 of Hs): rows kk, kk+1, col n = m
        bm.x = hs[kk * HPITCH + m];
        bm.y = hs[(kk + 1) * HPITCH + m];
        acc = __builtin_amdgcn_wmma_f32_16x16x4_f32(false, a, false, bm,
                                                    (short)0, acc, false, false);
    }

    // ---- Write tile: out = -O, D layout rows g / g+8 ----
    float* oplane = out + (size_t)b * HH * WW * CC + c;
    #pragma unroll
    for (int g = 0; g < 8; ++g) {
        int r = g + hi * 8;
        oplane[((size_t)(y0 + r) * WW + (x0 + m)) * CC] = -acc[g];
    }
}

extern "C" void kernel_launch(void* const* d_in, const int* in_sizes, int n_in,
                              void* d_out, int out_size, void* d_ws, size_t ws_size,
                              hipStream_t stream) {
    (void)in_sizes; (void)n_in; (void)d_ws; (void)ws_size; (void)out_size;
    const float* x = (const float*)d_in[0];
    float* out = (float*)d_out;

    const int tiles = (HH / TILE) * (WW / TILE);   // 32*32 = 1024
    dim3 grid(tiles, CC, 32);                      // (tile, channel, batch)
    dim3 block(32);                                // one wave32 per tile
    entropy9x9_wmma_kernel<<<grid, block, 0, stream>>>(x, out);
}